// AllReduce_50646254354848
// MI455X (gfx1250) — compile-verified
//
#include <hip/hip_runtime.h>
#include <cstdint>

typedef float v4f __attribute__((ext_vector_type(4)));
typedef int   v4i __attribute__((ext_vector_type(4)));
typedef __attribute__((address_space(1))) v4i v4i_global;
typedef __attribute__((address_space(3))) v4i v4i_lds;

#ifndef __has_builtin
#define __has_builtin(x) 0
#endif

#if __has_builtin(__builtin_amdgcn_global_load_async_to_lds_b128)
#define AR_HAS_ASYNC_LDS 1
#else
#define AR_HAS_ASYNC_LDS 0
#endif

namespace {
constexpr int   kTP      = 8;
constexpr int   kT       = 4096;
constexpr int   kH       = 8192;
constexpr int   kThreads = 1024;               // 32 wave32 waves per block
constexpr int   kV4      = kH / 4;             // 2048 float4 per row
constexpr float kEps     = 1e-5f;
constexpr float kInvH    = 1.0f / (float)kH;
}

__global__ __launch_bounds__(kThreads)
void fused_allreduce_rmsnorm(const float* __restrict__ x,          // [TP, T, H]
                             const float* __restrict__ residual,   // [T, H]
                             const float* __restrict__ wgt,        // [H]
                             float* __restrict__ out,              // [T, H]
                             float* __restrict__ resid_out)        // [T, H]
{
    __shared__ v4f   lds_res[kV4];             // 32 KB residual-row staging
    __shared__ float lds_part[kThreads / 32 + 1];

    const int    row     = blockIdx.x;
    const int    tid     = threadIdx.x;
    const int    i0      = tid;
    const int    i1      = tid + kThreads;
    const size_t rowbase = (size_t)row * (size_t)kV4;

    const v4f* res_row = (const v4f*)residual + rowbase;

#if AR_HAS_ASYNC_LDS
    // CDNA5 async copy (ASYNCcnt): residual row -> LDS, overlapped with the
    // 8-rank streaming accumulation below. Each thread copies exactly the two
    // float4s it will consume, so a wave-local s_wait_asynccnt suffices.
    __builtin_amdgcn_global_load_async_to_lds_b128(
        (v4i_global*)(uintptr_t)(res_row + i0),
        (v4i_lds*)(uint32_t)(uintptr_t)&lds_res[i0],
        0, 0);
    __builtin_amdgcn_global_load_async_to_lds_b128(
        (v4i_global*)(uintptr_t)(res_row + i1),
        (v4i_lds*)(uint32_t)(uintptr_t)&lds_res[i1],
        0, 0);
#endif

    // Stream 8 rank partials: read-once 1.07 GB -> non-temporal b128 loads so
    // they don't thrash L2 (which should keep the reusable norm weights).
    const v4f* x4 = (const v4f*)x;
    v4f a0 = {0.f, 0.f, 0.f, 0.f};
    v4f a1 = {0.f, 0.f, 0.f, 0.f};
#pragma unroll
    for (int r = 0; r < kTP; ++r) {
        const v4f* p = x4 + ((size_t)r * (size_t)kT) * (size_t)kV4 + rowbase;
        a0 += __builtin_nontemporal_load(p + i0);
        a1 += __builtin_nontemporal_load(p + i1);
    }

    v4f r0, r1;
#if AR_HAS_ASYNC_LDS
#if __has_builtin(__builtin_amdgcn_s_wait_asynccnt)
    __builtin_amdgcn_s_wait_asynccnt(0);
#else
    asm volatile("s_wait_asynccnt 0" ::: "memory");
#endif
    r0 = lds_res[i0];
    r1 = lds_res[i1];
#else
    r0 = __builtin_nontemporal_load(res_row + i0);
    r1 = __builtin_nontemporal_load(res_row + i1);
#endif

    const v4f ro0 = a0 + r0;
    const v4f ro1 = a1 + r1;

    // residual_out: write-once, never re-read -> NT stores.
    v4f* ro_out = (v4f*)resid_out + rowbase;
    __builtin_nontemporal_store(ro0, ro_out + i0);
    __builtin_nontemporal_store(ro1, ro_out + i1);

    // RMS reduction over the hidden dim (wave32 shuffles + tiny LDS scratch).
    const v4f q  = ro0 * ro0 + ro1 * ro1;
    float     ss = q.x + q.y + q.z + q.w;
#pragma unroll
    for (int off = 16; off > 0; off >>= 1)
        ss += __shfl_xor(ss, off, 32);

    const int wave = tid >> 5;
    const int lane = tid & 31;
    if (lane == 0) lds_part[wave] = ss;
    __syncthreads();
    if (wave == 0) {
        float s = lds_part[lane];
#pragma unroll
        for (int off = 16; off > 0; off >>= 1)
            s += __shfl_xor(s, off, 32);
        if (lane == 0) lds_part[kThreads / 32] = rsqrtf(s * kInvH + kEps);
    }
    __syncthreads();
    const float scale = lds_part[kThreads / 32];

    // norm_weight is reused by all 4096 rows: regular (temporal) loads.
    const v4f* w4 = (const v4f*)wgt;
    const v4f  o0 = ro0 * (scale * w4[i0]);
    const v4f  o1 = ro1 * (scale * w4[i1]);

    v4f* o_out = (v4f*)out + rowbase;
    __builtin_nontemporal_store(o0, o_out + i0);
    __builtin_nontemporal_store(o1, o_out + i1);
}

extern "C" void kernel_launch(void* const* d_in, const int* in_sizes, int n_in,
                              void* d_out, int out_size, void* d_ws, size_t ws_size,
                              hipStream_t stream) {
    (void)in_sizes; (void)n_in; (void)out_size; (void)d_ws; (void)ws_size;
    const float* x         = (const float*)d_in[0];
    const float* residual  = (const float*)d_in[1];
    const float* wgt       = (const float*)d_in[2];
    float*       out       = (float*)d_out;                       // output first
    float*       resid_out = out + (size_t)kT * (size_t)kH;       // then residual_out

    fused_allreduce_rmsnorm<<<dim3(kT), dim3(kThreads), 0, stream>>>(
        x, residual, wgt, out, resid_out);
}